// GatedGCN_72859825209691
// MI455X (gfx1250) — compile-verified
//
#include <hip/hip_runtime.h>

typedef __attribute__((ext_vector_type(2))) float v2f;
typedef __attribute__((ext_vector_type(8))) float v8f;

#define HID    128
#define HID3   384
#define OUTC   40
#define OUTCP  48   // OUT_C padded to a multiple of 16

// ---------------------------------------------------------------------------
// WMMA fp32 GEMM:  C[M,N] = op(A[M,128]) @ B (+bias), N compile-time constant
// (multiple of 16) so every address is base + immediate offset.
//   TRANSB=0: B is [128,N] row-major
//   TRANSB=1: B is [N,128] row-major (compute A @ B^T; contiguous b64 loads)
//   RELUA =1: relu applied to A on load (fused relu(h) @ W_out)
// One wave per 16-row tile. A row-tile preloaded once into 32 v2f fragments,
// reused across all N/16 column tiles. Per column tile: load all 32 B frags
// (one load clause), then a back-to-back chain of 32 v_wmma_f32_16x16x4_f32.
// Fragment layouts per CDNA5 ISA 7.12.2:
//   A(m,k): lane = m + 16*(k>>1), vgpr = k&1
//   B(k,n): lane = n + 16*(k>>1), vgpr = k&1
//   C(m,n): vgpr v -> m = v + 8*(lane>>4), n = lane&15
// ---------------------------------------------------------------------------
template <int N, int TRANSB, int RELUA>
__global__ void wmma_gemm_k128(const float* __restrict__ A,
                               const float* __restrict__ B,
                               const float* __restrict__ bias,
                               float* __restrict__ C,
                               int M)
{
    const int wave   = (blockIdx.x * blockDim.x + threadIdx.x) >> 5;
    const int mtiles = M >> 4;                 // M is a multiple of 16
    if (wave >= mtiles) return;                // wave-uniform exit

    const int lane = threadIdx.x & 31;
    const int lg   = lane >> 4;                // lane group (0: 0-15, 1: 16-31)
    const int ml   = lane & 15;                // m for A-frag, n for B/C-frag
    const int row0 = wave << 4;

    // Preload A row-tile (16x128): afrag[s] covers K = 4s..4s+3
    v2f afrag[32];
    const float* Arow = A + (size_t)(row0 + ml) * HID + 2 * lg;
#pragma unroll
    for (int s = 0; s < 32; ++s) {
        float ax = Arow[4 * s];
        float ay = Arow[4 * s + 1];
        if (RELUA) { ax = fmaxf(ax, 0.0f); ay = fmaxf(ay, 0.0f); }
        afrag[s].x = ax;
        afrag[s].y = ay;
    }

    const int ntiles = N >> 4;
    for (int nt = 0; nt < ntiles; ++nt) {
        const int col = (nt << 4) + ml;

        // Phase 1: load the whole 128(K) x 16(N) B tile -> one load clause
        v2f bfrag[32];
        if (TRANSB) {
            const float* Bp = B + (size_t)col * HID + 2 * lg;   // B[col][k]
#pragma unroll
            for (int s = 0; s < 32; ++s) {
                bfrag[s].x = Bp[4 * s];        // contiguous pair -> b64 load
                bfrag[s].y = Bp[4 * s + 1];
            }
        } else {
            const float* Bp = B + (size_t)(2 * lg) * N + col;   // B[k][col]
#pragma unroll
            for (int s = 0; s < 32; ++s) {
                bfrag[s].x = Bp[(4 * s) * N];  // immediate offsets (N is const)
                bfrag[s].y = Bp[(4 * s + 1) * N];
            }
        }

        // Phase 2: 32 back-to-back WMMAs accumulating into acc
        v8f acc = {0.f, 0.f, 0.f, 0.f, 0.f, 0.f, 0.f, 0.f};
#pragma unroll
        for (int s = 0; s < 32; ++s) {
            // (neg_a, A, neg_b, B, c_mod, C, reuse_a, reuse_b)
            acc = __builtin_amdgcn_wmma_f32_16x16x4_f32(
                false, afrag[s], false, bfrag[s], (short)0, acc, false, false);
        }

        const float bv = bias ? bias[col] : 0.0f;
        float* Crow = C + (size_t)(row0 + 8 * lg) * N + col;
#pragma unroll
        for (int v = 0; v < 8; ++v)
            Crow[(size_t)v * N] = acc[v] + bv;
    }
}

// ---------------------------------------------------------------------------
__global__ void zero_f32(float* __restrict__ p, int n)
{
    const int i = blockIdx.x * blockDim.x + threadIdx.x;
    if (i < n) p[i] = 0.0f;
}

// Pad W_out [128,40] -> [128,48] and b_out [40] -> [48] with zeros.
__global__ void pad_wout(const float* __restrict__ W, const float* __restrict__ b,
                         float* __restrict__ Wp, float* __restrict__ bp)
{
    const int i = blockIdx.x * blockDim.x + threadIdx.x;
    if (i < HID * OUTCP) {
        const int r = i / OUTCP;
        const int c = i - r * OUTCP;
        Wp[i] = (c < OUTC) ? W[r * OUTC + c] : 0.0f;
    }
    if (i < OUTCP)
        bp[i] = (i < OUTC) ? b[i] : 0.0f;
}

// One thread per (edge, 4-feature chunk): agg[dst] += m[src]
__global__ void edge_scatter_add(const float* __restrict__ m,
                                 const int* __restrict__ src,
                                 const int* __restrict__ dst,
                                 float* __restrict__ agg,
                                 int E)
{
    const int tid = blockIdx.x * blockDim.x + threadIdx.x;
    const int e   = tid >> 5;               // 32 float4 chunks cover HID=128
    if (e >= E) return;
    const int c4 = (tid & 31) << 2;
    const int s  = src[e];
    const int d  = dst[e];
    const float4 v = *reinterpret_cast<const float4*>(m + (size_t)s * HID + c4);
    float* a = agg + (size_t)d * HID + c4;
    atomicAdd(a + 0, v.x);
    atomicAdd(a + 1, v.y);
    atomicAdd(a + 2, v.z);
    atomicAdd(a + 3, v.w);
}

// GRU cell elementwise: hnew = (1-z)*tanh(i_n + r*h_n) + z*h
__global__ void gru_cell(const float* __restrict__ gi,
                         const float* __restrict__ gh,
                         const float* __restrict__ h,
                         float* __restrict__ hnew,
                         int N)
{
    const int idx = blockIdx.x * blockDim.x + threadIdx.x;
    if (idx >= N * HID) return;
    const int n = idx / HID;
    const int c = idx - n * HID;
    const float* gin = gi + (size_t)n * HID3;
    const float* ghn = gh + (size_t)n * HID3;
    const float r  = 1.0f / (1.0f + expf(-(gin[c]           + ghn[c])));
    const float z  = 1.0f / (1.0f + expf(-(gin[c +     HID] + ghn[c +     HID])));
    const float nn = tanhf(gin[c + 2 * HID] + r * ghn[c + 2 * HID]);
    hnew[idx] = (1.0f - z) * nn + z * h[idx];
}

// log-softmax: read 40 logits from padded stride-48 buffer, write stride-40.
__global__ void log_softmax_rows(const float* __restrict__ logits,
                                 float* __restrict__ out, int N)
{
    const int n = blockIdx.x * blockDim.x + threadIdx.x;
    if (n >= N) return;
    const float* row = logits + (size_t)n * OUTCP;
    float v[OUTC];
    float mx = -INFINITY;
    for (int c = 0; c < OUTC; ++c) { v[c] = row[c]; mx = fmaxf(mx, v[c]); }
    float s = 0.0f;
    for (int c = 0; c < OUTC; ++c) s += expf(v[c] - mx);
    const float lse = mx + logf(s);
    float* orow = out + (size_t)n * OUTC;
    for (int c = 0; c < OUTC; ++c) orow[c] = v[c] - lse;
}

// ---------------------------------------------------------------------------
extern "C" void kernel_launch(void* const* d_in, const int* in_sizes, int n_in,
                              void* d_out, int out_size, void* d_ws, size_t ws_size,
                              hipStream_t stream)
{
    (void)n_in; (void)out_size; (void)ws_size;

    const float* x       = (const float*)d_in[0];
    const int*   ei      = (const int*)  d_in[1];
    const float* W_in    = (const float*)d_in[2];
    const float* b_in    = (const float*)d_in[3];
    const float* layer_W = (const float*)d_in[4];
    const float* W_ih    = (const float*)d_in[5];
    const float* W_hh    = (const float*)d_in[6];
    const float* b_ih    = (const float*)d_in[7];
    const float* b_hh    = (const float*)d_in[8];
    const float* W_out   = (const float*)d_in[9];
    const float* b_out   = (const float*)d_in[10];
    float*       out     = (float*)d_out;

    const int N = in_sizes[0] / HID;              // 50000 (multiple of 16)
    const int E = in_sizes[1] / 2;                // 800000
    const int L = in_sizes[4] / (HID * HID);      // 3

    const int* e_src = ei;
    const int* e_dst = ei + E;

    // Workspace carve-up
    char* ws = (char*)d_ws;
    const size_t szH = (size_t)N * HID  * sizeof(float);
    const size_t szG = (size_t)N * HID3 * sizeof(float);
    const size_t szO = (size_t)N * OUTCP * sizeof(float);
    float* hA   = (float*)ws; ws += szH;
    float* hB   = (float*)ws; ws += szH;
    float* m    = (float*)ws; ws += szH;
    float* agg  = (float*)ws; ws += szH;
    float* gi   = (float*)ws; ws += szG;
    float* gh   = (float*)ws; ws += szG;
    float* lgts = (float*)ws; ws += szO;
    float* Wop  = (float*)ws; ws += (size_t)HID * OUTCP * sizeof(float);
    float* bop  = (float*)ws; ws += (size_t)OUTCP * sizeof(float);

    const int BLK        = 256;
    const int mtiles     = N >> 4;
    const int gemmBlocks = (mtiles * 32 + BLK - 1) / BLK;
    const int zeroBlocks = (N * HID + BLK - 1) / BLK;
    const int scatBlocks = (E * 32 + BLK - 1) / BLK;
    const int gruBlocks  = (N * HID + BLK - 1) / BLK;
    const int lsmBlocks  = (N + BLK - 1) / BLK;
    const int padBlocks  = (HID * OUTCP + BLK - 1) / BLK;

    // zero-padded copies of W_out / b_out (deterministic every call)
    pad_wout<<<padBlocks, BLK, 0, stream>>>(W_out, b_out, Wop, bop);

    // h = x @ W_in + b_in
    wmma_gemm_k128<HID, 0, 0><<<gemmBlocks, BLK, 0, stream>>>(x, W_in, b_in, hA, N);

    float* hcur  = hA;
    float* hnext = hB;
    for (int l = 0; l < L; ++l) {
        // m = h @ layer_W[l]
        wmma_gemm_k128<HID, 0, 0><<<gemmBlocks, BLK, 0, stream>>>(
            hcur, layer_W + (size_t)l * HID * HID, nullptr, m, N);
        // agg = segment_sum(m[src], dst)
        zero_f32<<<zeroBlocks, BLK, 0, stream>>>(agg, N * HID);
        edge_scatter_add<<<scatBlocks, BLK, 0, stream>>>(m, e_src, e_dst, agg, E);
        // gi = agg @ W_ih^T + b_ih ; gh = h @ W_hh^T + b_hh
        wmma_gemm_k128<HID3, 1, 0><<<gemmBlocks, BLK, 0, stream>>>(agg,  W_ih, b_ih, gi, N);
        wmma_gemm_k128<HID3, 1, 0><<<gemmBlocks, BLK, 0, stream>>>(hcur, W_hh, b_hh, gh, N);
        // GRU update
        gru_cell<<<gruBlocks, BLK, 0, stream>>>(gi, gh, hcur, hnext, N);
        float* t = hcur; hcur = hnext; hnext = t;
    }

    // logits(padded 48) = relu(h) @ Wop + bop ; then log_softmax -> d_out
    wmma_gemm_k128<OUTCP, 0, 1><<<gemmBlocks, BLK, 0, stream>>>(hcur, Wop, bop, lgts, N);
    log_softmax_rows<<<lsmBlocks, BLK, 0, stream>>>(lgts, out, N);
}